// _multiCodebookQuantization_6923487282645
// MI455X (gfx1250) — compile-verified
//
#include <hip/hip_runtime.h>

// ---------------------------------------------------------------------------
// Multi-codebook gumbel-softmax quantization for MI455X (gfx1250, wave32).
// Shapes: n=8, m=4, k=4096, d=128, h=w=16  -> 8192 pixels, 4096 codes, d=128.
// Outputs concatenated in d_out (fp32): sample | code | one_hot_code | logit.
// ---------------------------------------------------------------------------

typedef __attribute__((ext_vector_type(2))) float v2f;
typedef __attribute__((ext_vector_type(8))) float v8f;

#define K_     4096
#define D_     128
#define HW_    256        // h*w pixels per (n,m) image
#define PIX_   8192       // n*m*h*w
#define LDSP   132        // padded LDS row stride (floats): 16B-aligned, bank-shifted

// ---------------------------------------------------------------------------
// Kernel 1: inter = X . C^T via V_WMMA_F32_16X16X4_F32, fused dist->logit.
// Block = 128 threads = 4 waves. Block tile: 16 pixels x 64 codes.
// Each wave owns one 16(M=pixels) x 16(N=codes) WMMA tile, K-loop over d=128.
// ---------------------------------------------------------------------------
__global__ __launch_bounds__(128)
void mcq_logit_kernel(const float* __restrict__ x,
                      const float* __restrict__ codebook,
                      const float* __restrict__ temperature,
                      float* __restrict__ out_logit) {
  __shared__ float lds_a[16 * LDSP];       // 16 pixels x 128 d (padded)
  __shared__ float lds_b[4][16 * LDSP];    // per-wave: 16 codes x 128 d
  __shared__ float lds_x2[16];
  __shared__ float lds_c2[4][16];

  const int t    = threadIdx.x;
  const int lane = t & 31;
  const int wave = t >> 5;

  const int rowBase = blockIdx.y * 16;     // global pixel-row base
  const int nm   = rowBase >> 8;           // which (n,m) image (256 pix each)
  const int m_   = nm & 3;
  const int p0   = rowBase & 255;          // pixel offset inside the image
  const int kbase = blockIdx.x * 64 + wave * 16;

  // ---- stage A tile: x flat layout is (nm*128 + d)*256 + pixel ----
  const float* xb = x + (size_t)nm * (D_ * HW_) + p0;
  #pragma unroll
  for (int i = 0; i < 4; ++i) {
    int li = t + i * 128;                  // 0..511 float4-granules over (d, r4)
    int r4 = (li & 3) * 4;                 // pixel sub-row 0,4,8,12
    int dd = li >> 2;                      // d index 0..127
    const float4 v = *reinterpret_cast<const float4*>(xb + dd * HW_ + r4);
    lds_a[(r4 + 0) * LDSP + dd] = v.x;
    lds_a[(r4 + 1) * LDSP + dd] = v.y;
    lds_a[(r4 + 2) * LDSP + dd] = v.z;
    lds_a[(r4 + 3) * LDSP + dd] = v.w;
  }

  // ---- stage B tile: codebook[m_][kbase+j][0..127], d-contiguous ----
  const float* cbp = codebook + ((size_t)m_ * K_ + kbase) * D_;
  #pragma unroll
  for (int j = 0; j < 16; ++j) {
    const float4 v = *reinterpret_cast<const float4*>(cbp + j * D_ + lane * 4);
    *reinterpret_cast<float4*>(&lds_b[wave][j * LDSP + lane * 4]) = v;
  }
  __syncthreads();

  // ---- row norms: ||x||^2 per pixel, ||c||^2 per code ----
  if (wave == 0 && lane < 16) {
    float s = 0.f;
    #pragma unroll 8
    for (int dd = 0; dd < D_; ++dd) {
      float a = lds_a[lane * LDSP + dd];
      s += a * a;
    }
    lds_x2[lane] = s;
  }
  if (lane < 16) {
    float s = 0.f;
    #pragma unroll 8
    for (int dd = 0; dd < D_; ++dd) {
      float b = lds_b[wave][lane * LDSP + dd];
      s += b * b;
    }
    lds_c2[wave][lane] = s;
  }
  __syncthreads();

  // ---- GEMM: 32 steps of V_WMMA_F32_16X16X4_F32 ----
  // A 16x4 layout: lanes 0-15 hold M=lane, K={k0,k0+1}; lanes 16-31 K={k0+2,k0+3}.
  // B 4x16 layout: lanes 0-15 hold N=lane, K={k0,k0+1}; lanes 16-31 K={k0+2,k0+3}.
  const int half = (lane < 16) ? 0 : 2;
  const int rsel = (lane & 15) * LDSP;
  v8f acc = {};
  #pragma unroll 8
  for (int k0 = 0; k0 < D_; k0 += 4) {
    v2f a = *reinterpret_cast<const v2f*>(&lds_a[rsel + k0 + half]);
    v2f b = *reinterpret_cast<const v2f*>(&lds_b[wave][rsel + k0 + half]);
    acc = __builtin_amdgcn_wmma_f32_16x16x4_f32(false, a, false, b,
                                                (short)0, acc, false, false);
  }

  // ---- epilogue: logit = -(x2 + c2 - 2*inter) * max(temp,eps) / sqrt(k) ----
  float tm = temperature[m_];
  tm = (tm < 1e-6f) ? 1e-6f : tm;
  const float nscale = -tm * (1.0f / 64.0f);     // -temp / sqrt(4096)
  const int   N      = lane & 15;
  const float c2v    = lds_c2[wave][N];
  const int   mofs   = (lane < 16) ? 0 : 8;      // C layout: vgpr r -> M = r (+8)
  float* outp = out_logit + (size_t)rowBase * K_ + kbase + N;
  #pragma unroll
  for (int r = 0; r < 8; ++r) {
    int M = r + mofs;
    float dist = lds_x2[M] + c2v - 2.0f * acc[r];
    outp[(size_t)M * K_] = dist * nscale;
  }
}

// ---------------------------------------------------------------------------
// Kernel 2: per-pixel argmax over k=4096 of logit (-> code/one_hot) and of
// logit + gumbel (-> sample).  One wave per pixel; logit re-read hits L2.
// ---------------------------------------------------------------------------
__global__ __launch_bounds__(128)
void mcq_argmax_kernel(const float* __restrict__ logit,
                       const float* __restrict__ noise,
                       float* __restrict__ out_sample,
                       float* __restrict__ out_code,
                       float* __restrict__ out_onehot) {
  const int lane = threadIdx.x & 31;
  const int wave = threadIdx.x >> 5;
  const int P = blockIdx.x * 4 + wave;

  const float4* lg4 = reinterpret_cast<const float4*>(logit + (size_t)P * K_);
  const float4* nz4 = reinterpret_cast<const float4*>(noise + (size_t)P * K_);

  float bc = -3.0e38f; int bci = 0;   // clean argmax  -> code / one_hot_code
  float bg = -3.0e38f; int bgi = 0;   // perturbed     -> sample (hard one-hot)
  for (int c = 0; c < 32; ++c) {
    int vi = c * 32 + lane;           // ascending kk per lane -> first-max wins
    float4 l = lg4[vi];
    float4 u = nz4[vi];
    float lv[4] = {l.x, l.y, l.z, l.w};
    float uv[4] = {u.x, u.y, u.z, u.w};
    int kk = vi * 4;
    #pragma unroll
    for (int j = 0; j < 4; ++j) {
      float lj = lv[j];
      if (lj > bc) { bc = lj; bci = kk + j; }
      float g = lj - __logf(-__logf(uv[j]));   // logit + gumbel, TAU = 1
      if (g > bg) { bg = g; bgi = kk + j; }
    }
  }
  // wave32 reduction; prefer smaller index on exact ties (jnp.argmax semantics)
  #pragma unroll
  for (int off = 16; off >= 1; off >>= 1) {
    float oc = __shfl_xor(bc, off); int oci = __shfl_xor(bci, off);
    if (oc > bc || (oc == bc && oci < bci)) { bc = oc; bci = oci; }
    float og = __shfl_xor(bg, off); int ogi = __shfl_xor(bgi, off);
    if (og > bg || (og == bg && ogi < bgi)) { bg = og; bgi = ogi; }
  }
  bci = __shfl(bci, 0);
  bgi = __shfl(bgi, 0);
  if (lane == 0) out_code[P] = (float)bci;

  float4* sm4 = reinterpret_cast<float4*>(out_sample + (size_t)P * K_);
  float4* oh4 = reinterpret_cast<float4*>(out_onehot + (size_t)P * K_);
  for (int c = 0; c < 32; ++c) {
    int vi = c * 32 + lane;
    int kk = vi * 4;
    float4 s, o;
    s.x = (kk + 0 == bgi) ? 1.0f : 0.0f;
    s.y = (kk + 1 == bgi) ? 1.0f : 0.0f;
    s.z = (kk + 2 == bgi) ? 1.0f : 0.0f;
    s.w = (kk + 3 == bgi) ? 1.0f : 0.0f;
    o.x = (kk + 0 == bci) ? 1.0f : 0.0f;
    o.y = (kk + 1 == bci) ? 1.0f : 0.0f;
    o.z = (kk + 2 == bci) ? 1.0f : 0.0f;
    o.w = (kk + 3 == bci) ? 1.0f : 0.0f;
    sm4[vi] = s;
    oh4[vi] = o;
  }
}

// ---------------------------------------------------------------------------
extern "C" void kernel_launch(void* const* d_in, const int* in_sizes, int n_in,
                              void* d_out, int out_size, void* d_ws, size_t ws_size,
                              hipStream_t stream) {
  const float* x     = (const float*)d_in[0];   // (8, 512, 16, 16)
  const float* cb    = (const float*)d_in[1];   // (4, 4096, 128)
  const float* temp  = (const float*)d_in[2];   // (4,)
  const float* noise = (const float*)d_in[3];   // (8, 4, 16, 16, 4096)

  float* out        = (float*)d_out;            // concat: sample|code|one_hot|logit
  float* out_sample = out;
  float* out_code   = out_sample + (size_t)PIX_ * K_;
  float* out_onehot = out_code + PIX_;
  float* out_logit  = out_onehot + (size_t)PIX_ * K_;

  dim3 g1(K_ / 64, PIX_ / 16);                  // 64 x 512 blocks
  mcq_logit_kernel<<<g1, 128, 0, stream>>>(x, cb, temp, out_logit);

  mcq_argmax_kernel<<<PIX_ / 4, 128, 0, stream>>>(out_logit, noise,
                                                  out_sample, out_code, out_onehot);
}